// LinearAttnFFN_24747601560089
// MI455X (gfx1250) — compile-verified
//
#include <hip/hip_runtime.h>
#include <math.h>

// ---------------------------------------------------------------------------
// LinearAttnFFN block for MI455X (gfx1250), fp32 via V_WMMA_F32_16X16X4_F32.
//   B=32, C=256, P=4, N=1024, S=P*N=4096, F=512.
// Five kernels on one stream; ws holds qkv/x1/h intermediates (~385 MB).
// GEMMs are register-blocked: one A (weight) fragment feeds NB WMMAs.
// ---------------------------------------------------------------------------

#define B_DIM 32
#define C_DIM 256
#define F_DIM 512
#define P_DIM 4
#define N_DIM 1024
#define S_DIM 4096            // P*N
#define EPS   1e-5f

#define TILE  64              // positions per block (K1/K3/K4)
#define PITCH (C_DIM + 2)     // LDS pitch: [pos][chan], 258 floats
#define TILE5 32              // positions per block (K5, K=512)
#define PITCH5 (F_DIM + 2)    // 514 floats

typedef __attribute__((ext_vector_type(2))) float v2f;
typedef __attribute__((ext_vector_type(8))) float v8f;

// NB 16x16 output tiles sharing one A (weight) fragment stream.
// Per-lane: m = lane&15 selects A row / B column, kk = (lane>>4)*2 selects
// the K-pair per the CDNA5 f32 A/B VGPR striping (ISA 7.12.2).
template <int NB>
__device__ __forceinline__ void wmma_k_loop_nb(const float* __restrict__ Wr,
                                               const float* __restrict__ Ybase,
                                               int pitch, int K, int kk, int m,
                                               v8f acc[NB]) {
#pragma unroll 4
  for (int k = 0; k < K; k += 4) {
    v2f a = *(const v2f*)(Wr + k + kk);      // W[m][k+kk], W[m][k+kk+1]
#pragma unroll
    for (int c = 0; c < NB; ++c) {
      v2f b = *(const v2f*)(Ybase + (c * 16 + m) * pitch + k + kk);
      acc[c] = __builtin_amdgcn_wmma_f32_16x16x4_f32(
          /*neg_a=*/false, a, /*neg_b=*/false, b,
          /*c_mod=*/(short)0, acc[c], /*reuse_a=*/false, /*reuse_b=*/false);
    }
  }
}

// ---------------------------------------------------------------------------
// K1 / K4: fused channel-LayerNorm + (512 x 256) GEMM.
//   WITH_Q: also emit q row (weight row 0), weight rows shifted by 1.
//   LEAKY : leaky-ReLU(0.1) epilogue (ffn1).
// Block: 256 threads, tile = 64 positions x 256 channels in LDS (transposed).
// ---------------------------------------------------------------------------
template <bool WITH_Q, bool LEAKY>
__global__ void ln_gemm_kernel(const float* __restrict__ xin,
                               const float* __restrict__ lnw,
                               const float* __restrict__ lnb,
                               const float* __restrict__ W,    // rows x 256
                               const float* __restrict__ bias,
                               float* __restrict__ out512,     // (B,512,S)
                               float* __restrict__ qout) {     // (B,S) | null
  extern __shared__ float Y[];               // [TILE][PITCH]
  __shared__ float red[4][TILE];
  __shared__ float red2[4][TILE];
  __shared__ float meanv[TILE];
  __shared__ float rstdv[TILE];

  const int b  = blockIdx.y;
  const int s0 = blockIdx.x * TILE;
  const int t  = threadIdx.x;
  const int j  = t & (TILE - 1);             // position within tile
  const int ch = t >> 6;                     // channel chunk 0..3
  const int c0 = ch * 64;

  // Phase 1: coalesced load of the 256x64 tile + LN partial sums.
  const float* xb = xin + (size_t)b * C_DIM * S_DIM + s0 + j;
  float sum = 0.f, sq = 0.f;
#pragma unroll 4
  for (int i = 0; i < 64; ++i) {
    int c = c0 + i;
    float v = xb[(size_t)c * S_DIM];
    Y[j * PITCH + c] = v;
    sum += v;
    sq += v * v;
  }
  red[ch][j] = sum;
  red2[ch][j] = sq;
  __syncthreads();
  if (t < TILE) {
    float s  = red[0][t] + red[1][t] + red[2][t] + red[3][t];
    float q2 = red2[0][t] + red2[1][t] + red2[2][t] + red2[3][t];
    float mean = s * (1.f / C_DIM);
    float var  = q2 * (1.f / C_DIM) - mean * mean;
    meanv[t] = mean;
    rstdv[t] = __frsqrt_rn(var + EPS);
  }
  __syncthreads();

  // Phase 2: normalize in LDS.
  {
    float mean = meanv[j], rstd = rstdv[j];
#pragma unroll 4
    for (int i = 0; i < 64; ++i) {
      int c = c0 + i;
      Y[j * PITCH + c] = (Y[j * PITCH + c] - mean) * rstd * lnw[c] + lnb[c];
    }
  }
  __syncthreads();

  // Phase 3 (qkv only): q row = W[0] . y  per position.
  if (WITH_Q) {
    float qs = 0.f;
#pragma unroll 4
    for (int i = 0; i < 64; ++i) {
      int c = c0 + i;
      qs += W[c] * Y[j * PITCH + c];
    }
    red[ch][j] = qs;
    __syncthreads();
    if (t < TILE) {
      qout[(size_t)b * S_DIM + s0 + t] =
          red[0][t] + red[1][t] + red[2][t] + red[3][t] + bias[0];
    }
    __syncthreads();
  }

  // Phase 4: WMMA GEMM. 32 row-tiles / 8 waves; 4 position-tiles blocked in
  // registers so each weight fragment feeds 4 WMMAs.
  const int wave = t >> 5;
  const int lane = t & 31;
  const int m    = lane & 15;                // A row / B col / D col
  const int kk   = (lane >> 4) << 1;         // K-pair select (0 or 2)
  const int mh   = (lane >> 4) << 3;         // D: vgpr r -> M = mh + r
  const int rofs = WITH_Q ? 1 : 0;

  for (int rt = wave; rt < 32; rt += 8) {
    const float* Wr = W + (size_t)(rofs + rt * 16 + m) * C_DIM;
    __builtin_prefetch(Wr, 0, 1);            // global_prefetch_b8
    v8f acc[4] = {};
    wmma_k_loop_nb<4>(Wr, Y, PITCH, C_DIM, kk, m, acc);
#pragma unroll
    for (int c = 0; c < 4; ++c) {
#pragma unroll
      for (int r = 0; r < 8; ++r) {
        int row = rt * 16 + mh + r;
        float d = acc[c][r] + bias[rofs + row];
        if (LEAKY) d = d > 0.f ? d : 0.1f * d;
        out512[((size_t)b * F_DIM + row) * S_DIM + s0 + c * 16 + m] = d;
      }
    }
  }
}

// ---------------------------------------------------------------------------
// K2: softmax over n (per b,p) + ctx[c] = sum_n k[c,n] * p[n].
// One block per (b,p); 256 threads.
// ---------------------------------------------------------------------------
__global__ void softmax_ctx_kernel(const float* __restrict__ qbuf,
                                   const float* __restrict__ kv,
                                   float* __restrict__ ctx) {
  __shared__ float probs[N_DIM];
  __shared__ float red[256];
  const int bp = blockIdx.x;                 // 0..127
  const int b = bp >> 2, p = bp & 3;
  const int t = threadIdx.x;

  const float* q = qbuf + (size_t)b * S_DIM + p * N_DIM;
  float v[4];
  float mx = -INFINITY;
#pragma unroll
  for (int i = 0; i < 4; ++i) {
    v[i] = q[t + i * 256];
    mx = fmaxf(mx, v[i]);
  }
  red[t] = mx;
  __syncthreads();
  for (int o = 128; o > 0; o >>= 1) {
    if (t < o) red[t] = fmaxf(red[t], red[t + o]);
    __syncthreads();
  }
  mx = red[0];
  __syncthreads();

  float s = 0.f;
#pragma unroll
  for (int i = 0; i < 4; ++i) {
    v[i] = __expf(v[i] - mx);
    s += v[i];
  }
  red[t] = s;
  __syncthreads();
  for (int o = 128; o > 0; o >>= 1) {
    if (t < o) red[t] += red[t + o];
    __syncthreads();
  }
  float inv = 1.f / red[0];
#pragma unroll
  for (int i = 0; i < 4; ++i) probs[t + i * 256] = v[i] * inv;
  __syncthreads();

  // ctx: thread t owns channel t; stream k row with float4 loads.
  const float* kr = kv + ((size_t)b * F_DIM + t) * S_DIM + p * N_DIM;
  float acc = 0.f;
#pragma unroll 4
  for (int n = 0; n < N_DIM; n += 4) {
    float4 kx = *reinterpret_cast<const float4*>(kr + n);
    acc += kx.x * probs[n] + kx.y * probs[n + 1] +
           kx.z * probs[n + 2] + kx.w * probs[n + 3];
  }
  ctx[(size_t)bp * C_DIM + t] = acc;
}

// ---------------------------------------------------------------------------
// K3: A = relu(v) * ctx ; x1 = x + out_w @ A + out_b.
// ---------------------------------------------------------------------------
__global__ void attn_out_kernel(const float* __restrict__ x,
                                const float* __restrict__ kv,
                                const float* __restrict__ ctx,
                                const float* __restrict__ W,   // 256x256
                                const float* __restrict__ bias,
                                float* __restrict__ x1) {
  extern __shared__ float A[];               // [TILE][PITCH]
  const int b  = blockIdx.y;
  const int s0 = blockIdx.x * TILE;
  const int p  = s0 / N_DIM;
  const int t  = threadIdx.x;
  const int j  = t & (TILE - 1);
  const int ch = t >> 6;
  const int c0 = ch * 64;

  const float* ctxbp = ctx + ((size_t)b * P_DIM + p) * C_DIM;
#pragma unroll 4
  for (int i = 0; i < 64; ++i) {
    int c = c0 + i;
    float vv = kv[((size_t)b * F_DIM + C_DIM + c) * S_DIM + s0 + j];
    vv = vv > 0.f ? vv : 0.f;
    A[j * PITCH + c] = vv * ctxbp[c];
  }
  __syncthreads();

  const int wave = t >> 5;
  const int lane = t & 31;
  const int m    = lane & 15;
  const int kk   = (lane >> 4) << 1;
  const int mh   = (lane >> 4) << 3;

  for (int rt = wave; rt < 16; rt += 8) {    // 2 row-tiles per wave
    const float* Wr = W + (size_t)(rt * 16 + m) * C_DIM;
    __builtin_prefetch(Wr, 0, 1);
    v8f acc[4] = {};
    wmma_k_loop_nb<4>(Wr, A, PITCH, C_DIM, kk, m, acc);
#pragma unroll
    for (int c = 0; c < 4; ++c) {
#pragma unroll
      for (int r = 0; r < 8; ++r) {
        int row = rt * 16 + mh + r;
        size_t idx = ((size_t)b * C_DIM + row) * S_DIM + s0 + c * 16 + m;
        x1[idx] = x[idx] + bias[row] + acc[c][r];
      }
    }
  }
}

// ---------------------------------------------------------------------------
// K5: out = x1 + ffn2_w(256x512) @ h + ffn2_b.  Tile = 32 positions, K=512.
// ---------------------------------------------------------------------------
__global__ void ffn2_kernel(const float* __restrict__ x1,
                            const float* __restrict__ h,
                            const float* __restrict__ W,       // 256x512
                            const float* __restrict__ bias,
                            float* __restrict__ out) {
  extern __shared__ float Hs[];              // [TILE5][PITCH5]
  const int b  = blockIdx.y;
  const int s0 = blockIdx.x * TILE5;
  const int t  = threadIdx.x;
  const int j  = t & (TILE5 - 1);
  const int ch = t >> 5;                     // 0..7, 64 rows each
#pragma unroll 4
  for (int i = 0; i < 64; ++i) {
    int r = ch * 64 + i;
    Hs[j * PITCH5 + r] = h[((size_t)b * F_DIM + r) * S_DIM + s0 + j];
  }
  __syncthreads();

  const int wave = t >> 5;
  const int lane = t & 31;
  const int m    = lane & 15;
  const int kk   = (lane >> 4) << 1;
  const int mh   = (lane >> 4) << 3;

  for (int rt = wave; rt < 16; rt += 8) {    // 2 row-tiles per wave
    const float* Wr = W + (size_t)(rt * 16 + m) * F_DIM;
    __builtin_prefetch(Wr, 0, 1);
    v8f acc[2] = {};
    wmma_k_loop_nb<2>(Wr, Hs, PITCH5, F_DIM, kk, m, acc);
#pragma unroll
    for (int c = 0; c < 2; ++c) {
#pragma unroll
      for (int r = 0; r < 8; ++r) {
        int row = rt * 16 + mh + r;
        size_t idx = ((size_t)b * C_DIM + row) * S_DIM + s0 + c * 16 + m;
        out[idx] = x1[idx] + bias[row] + acc[c][r];
      }
    }
  }
}

// ---------------------------------------------------------------------------
extern "C" void kernel_launch(void* const* d_in, const int* in_sizes, int n_in,
                              void* d_out, int out_size, void* d_ws,
                              size_t ws_size, hipStream_t stream) {
  const float* x      = (const float*)d_in[0];
  const float* ln1_w  = (const float*)d_in[1];
  const float* ln1_b  = (const float*)d_in[2];
  const float* qkv_w  = (const float*)d_in[3];   // (513,256)
  const float* qkv_b  = (const float*)d_in[4];
  const float* out_w  = (const float*)d_in[5];   // (256,256)
  const float* out_b  = (const float*)d_in[6];
  const float* ln2_w  = (const float*)d_in[7];
  const float* ln2_b  = (const float*)d_in[8];
  const float* ffn1_w = (const float*)d_in[9];   // (512,256)
  const float* ffn1_b = (const float*)d_in[10];
  const float* ffn2_w = (const float*)d_in[11];  // (256,512)
  const float* ffn2_b = (const float*)d_in[12];
  float* outp = (float*)d_out;

  // Workspace layout (floats). kv buffer is reused for h after K3 (stream
  // execution is in-order, so the alias is safe).
  float* ws   = (float*)d_ws;
  float* qbuf = ws;                                   // B*S        = 131072
  float* ctxb = qbuf + (size_t)B_DIM * S_DIM;         // B*P*C      =  32768
  float* kvb  = ctxb + (size_t)B_DIM * P_DIM * C_DIM; // B*512*S    = 67108864
  float* x1   = kvb + (size_t)B_DIM * F_DIM * S_DIM;  // B*256*S    = 33554432
  float* hbuf = kvb;                                  // alias

  const size_t lds1 = (size_t)TILE * PITCH * sizeof(float);     // ~64.5 KB
  const size_t lds5 = (size_t)TILE5 * PITCH5 * sizeof(float);   // ~64.3 KB
  dim3 grid1(S_DIM / TILE, B_DIM);   // (64, 32)
  dim3 grid5(S_DIM / TILE5, B_DIM);  // (128, 32)

  // 1) LN1 + QKV  -> qbuf, kvb
  ln_gemm_kernel<true, false><<<grid1, 256, lds1, stream>>>(
      x, ln1_w, ln1_b, qkv_w, qkv_b, kvb, qbuf);
  // 2) softmax(q) + ctx = sum_n k*p
  softmax_ctx_kernel<<<B_DIM * P_DIM, 256, 0, stream>>>(qbuf, kvb, ctxb);
  // 3) x1 = x + out_w @ (relu(v)*ctx) + out_b
  attn_out_kernel<<<grid1, 256, lds1, stream>>>(
      x, kvb, ctxb, out_w, out_b, x1);
  // 4) h = leaky(ffn1 @ LN2(x1) + b)
  ln_gemm_kernel<false, true><<<grid1, 256, lds1, stream>>>(
      x1, ln2_w, ln2_b, ffn1_w, ffn1_b, hbuf, nullptr);
  // 5) out = x1 + ffn2 @ h + b
  ffn2_kernel<<<grid5, 256, lds5, stream>>>(
      x1, hbuf, ffn2_w, ffn2_b, outp);
}